// GatedGCN_27900107555155
// MI455X (gfx1250) — compile-verified
//
#include <hip/hip_runtime.h>
#include <hip/hip_bf16.h>
#include <math.h>

#define HD 128          // hidden width
#define LPAD 68         // 64-wide K chunk + 4 pad -> 8B aligned, conflict-free ds_load_b64
#define NLAYER 4

typedef __attribute__((ext_vector_type(2))) float v2f;
typedef __attribute__((ext_vector_type(8))) float v8f;

// Y[rows,128] = epilogue( X[rows,128] @ W[128,128] )
//   epilogue: +bias[col] (opt), +add[row,col] (opt),
//             if res: Y = res + relu(acc) else Y = acc
// One wave -> 16 rows x 128 cols (8 WMMA f32 16x16x4 tiles, K looped 4 at a time).
// Requires rows % 16 == 0 for full tiles (N = 50000 satisfies this).
__global__ __launch_bounds__(256) void gemm128_wmma(
    const float* __restrict__ X, const float* __restrict__ W,
    const float* __restrict__ bias, const float* __restrict__ add,
    const float* __restrict__ res, float* __restrict__ Y, int rows)
{
  __shared__ float Bs[HD * LPAD];   // Bs[n*LPAD + kr] = W[(kk+kr)*HD + n] (transposed chunk)
  const int tid   = threadIdx.x;
  const int wave  = tid >> 5;
  const int lane  = tid & 31;
  const int ln    = lane & 15;
  const int khalf = (lane >> 4) << 1;          // 0 for lanes 0-15, 2 for lanes 16-31
  const int row0  = blockIdx.x * 128 + wave * 16;

  v8f acc[8] = {};
  const float* arow = X + (size_t)(row0 + ln) * HD + khalf;

  for (int kk = 0; kk < HD; kk += 64) {
    __syncthreads();
    // stage 64 K-rows of W, transposed, coalesced global reads
    for (int i = tid; i < 64 * HD; i += 256) {
      int kr = i >> 7;          // 0..63
      int n  = i & 127;         // 0..127
      Bs[n * LPAD + kr] = W[(size_t)(kk + kr) * HD + n];
    }
    __syncthreads();
    if (row0 < rows) {
      for (int k = 0; k < 64; k += 4) {
        // A 16x4 fragment: lane<16 -> K=k,k+1 ; lane>=16 -> K=k+2,k+3 (ISA layout)
        v2f a = *(const v2f*)(arow + kk + k);
#pragma unroll
        for (int t = 0; t < 8; ++t) {
          // B 4x16 fragment for cols t*16..t*16+15, same K striping as A
          v2f b = *(const v2f*)&Bs[(t * 16 + ln) * LPAD + k + khalf];
          acc[t] = __builtin_amdgcn_wmma_f32_16x16x4_f32(
              false, a, false, b, (short)0, acc[t], false, false);
        }
      }
    }
  }

  if (row0 >= rows) return;
#pragma unroll
  for (int t = 0; t < 8; ++t) {
    const int col = t * 16 + ln;
    const float bv = bias ? bias[col] : 0.f;
#pragma unroll
    for (int v = 0; v < 8; ++v) {
      const int row = row0 + v + ((lane >> 4) << 3);  // C/D: lanes>=16 hold M=v+8
      float val = acc[t][v] + bv;
      const size_t off = (size_t)row * HD + col;
      if (add) val += add[off];
      if (res) val = res[off] + fmaxf(val, 0.f);
      Y[off] = val;
    }
  }
}

// One wave per edge: eta = sigmoid(gr[rec] + gs[send]); agg[rec] += eta * hs[send]
__global__ __launch_bounds__(256) void edge_kernel(
    const int* __restrict__ rec, const int* __restrict__ send,
    const float* __restrict__ gr, const float* __restrict__ gs,
    const float* __restrict__ hs, float* __restrict__ agg, int nedge)
{
  const int e = blockIdx.x * 8 + (threadIdx.x >> 5);
  if (e >= nedge) return;
  const int lane = threadIdx.x & 31;
  const int r = rec[e], s = send[e];
  const int c = lane << 2;
  const float4 a = *(const float4*)(gr + (size_t)r * HD + c);
  const float4 b = *(const float4*)(gs + (size_t)s * HD + c);
  const float4 m = *(const float4*)(hs + (size_t)s * HD + c);
  float* dst = agg + (size_t)r * HD + c;
  atomicAdd(dst + 0, m.x / (1.f + __expf(-(a.x + b.x))));
  atomicAdd(dst + 1, m.y / (1.f + __expf(-(a.y + b.y))));
  atomicAdd(dst + 2, m.z / (1.f + __expf(-(a.z + b.z))));
  atomicAdd(dst + 3, m.w / (1.f + __expf(-(a.w + b.w))));
}

__global__ void zero_kernel(float* __restrict__ p, long n) {
  long i = (long)blockIdx.x * blockDim.x + threadIdx.x;
  if (i < n) p[i] = 0.f;
}

// hagg[batch[i]] += h[i]  (float4 per thread)
__global__ __launch_bounds__(256) void pool_kernel(
    const float* __restrict__ h, const int* __restrict__ batch,
    float* __restrict__ hagg, int n)
{
  const int idx = blockIdx.x * 256 + threadIdx.x;
  const int i = idx >> 5;
  if (i >= n) return;
  const int c = (idx & 31) << 2;
  const int g = batch[i];
  const float4 v = *(const float4*)(h + (size_t)i * HD + c);
  float* dst = hagg + (size_t)g * HD + c;
  atomicAdd(dst + 0, v.x);
  atomicAdd(dst + 1, v.y);
  atomicAdd(dst + 2, v.z);
  atomicAdd(dst + 3, v.w);
}

// out[g] = relu(hagg[g] @ W1 + b1) @ W2 + b2 ; one wave per graph (H/2 = 64 hidden)
__global__ __launch_bounds__(256) void mlp_head(
    const float* __restrict__ hagg, const float* __restrict__ W1,
    const float* __restrict__ b1, const float* __restrict__ W2,
    const float* __restrict__ b2, float* __restrict__ out, int ngraph)
{
  const int g = blockIdx.x * 8 + (threadIdx.x >> 5);
  if (g >= ngraph) return;
  const int lane = threadIdx.x & 31;
  const float* hg = hagg + (size_t)g * HD;
  float t0 = b1[lane], t1 = b1[lane + 32];
  for (int k = 0; k < HD; ++k) {
    const float hv = hg[k];
    t0 += hv * W1[k * 64 + lane];
    t1 += hv * W1[k * 64 + lane + 32];
  }
  float p = fmaxf(t0, 0.f) * W2[lane] + fmaxf(t1, 0.f) * W2[lane + 32];
  for (int off = 16; off > 0; off >>= 1) p += __shfl_xor(p, off, 32);
  if (lane == 0) out[g] = p + b2[0];
}

extern "C" void kernel_launch(void* const* d_in, const int* in_sizes, int n_in,
                              void* d_out, int out_size, void* d_ws, size_t ws_size,
                              hipStream_t stream) {
  const float* h_in    = (const float*)d_in[0];
  const int*   eidx    = (const int*)d_in[1];
  const int*   batch   = (const int*)d_in[2];
  const float* W_embed = (const float*)d_in[3];
  const float* b_embed = (const float*)d_in[4];
  const float* Wg      = (const float*)d_in[5];   // [L][256][128]
  const float* bg      = (const float*)d_in[6];   // [L][128]
  const float* Wsm     = (const float*)d_in[7];   // [L][128][128]
  const float* bsm     = (const float*)d_in[8];
  const float* Wr      = (const float*)d_in[9];
  const float* br      = (const float*)d_in[10];
  const float* W1      = (const float*)d_in[11];  // [128][64]
  const float* b1      = (const float*)d_in[12];
  const float* W2      = (const float*)d_in[13];  // [64][1]
  const float* b2      = (const float*)d_in[14];
  float* out = (float*)d_out;

  const int N = in_sizes[0] / HD;
  const int E = in_sizes[1] / 2;
  const int G = out_size;
  const int* send = eidx;        // edge_index[0]
  const int* rec  = eidx + E;    // edge_index[1]

  // workspace layout
  float* p = (float*)d_ws;
  const size_t NH = (size_t)N * HD;
  float* hb0  = p; p += NH;
  float* hb1  = p; p += NH;
  float* grb  = p; p += NH;
  float* gsb  = p; p += NH;
  float* hsb  = p; p += NH;
  float* aggb = p; p += NH;
  float* hagg = p; p += (size_t)G * HD;
  (void)ws_size; (void)n_in;

  const int ggrid = (N + 127) / 128;     // 128 rows per block (8 waves x 16 rows)
  const int zgrid = (int)((NH + 255) / 256);

  // h = h_in @ W_embed + b_embed
  gemm128_wmma<<<ggrid, 256, 0, stream>>>(h_in, W_embed, b_embed, nullptr, nullptr, hb0, N);

  float* cur = hb0;
  float* nxt = hb1;
  for (int l = 0; l < NLAYER; ++l) {
    const float* Wg_l = Wg + (size_t)l * 2 * HD * HD;
    // gr = h @ Wg_top + bg ; gs = h @ Wg_bot ; hs = h @ Ws + bs
    gemm128_wmma<<<ggrid, 256, 0, stream>>>(cur, Wg_l, bg + l * HD, nullptr, nullptr, grb, N);
    gemm128_wmma<<<ggrid, 256, 0, stream>>>(cur, Wg_l + HD * HD, nullptr, nullptr, nullptr, gsb, N);
    gemm128_wmma<<<ggrid, 256, 0, stream>>>(cur, Wsm + (size_t)l * HD * HD, bsm + l * HD,
                                            nullptr, nullptr, hsb, N);
    zero_kernel<<<zgrid, 256, 0, stream>>>(aggb, (long)NH);
    edge_kernel<<<(E + 7) / 8, 256, 0, stream>>>(rec, send, grb, gsb, hsb, aggb, E);
    // h = h + relu(h @ Wr + br + agg)
    gemm128_wmma<<<ggrid, 256, 0, stream>>>(cur, Wr + (size_t)l * HD * HD, br + l * HD,
                                            aggb, cur, nxt, N);
    float* tmp = cur; cur = nxt; nxt = tmp;
  }

  zero_kernel<<<(G * HD + 255) / 256, 256, 0, stream>>>(hagg, (long)G * HD);
  pool_kernel<<<(N * 32 + 255) / 256, 256, 0, stream>>>(cur, batch, hagg, N);
  mlp_head<<<(G + 7) / 8, 256, 0, stream>>>(hagg, W1, b1, W2, b2, out, G);
}